// FallbackSLayerExponential_20486994002573
// MI455X (gfx1250) — compile-verified
//
#include <hip/hip_runtime.h>
#include <hip/hip_bf16.h>
#include <math.h>

// CDNA5 / gfx1250. wave32. WMMA f32 16x16x4, two chained steps => K=8 rank
// decomposition of the anisotropic squared distance:
//   q[n,e] = s0(p0-c0)^2 + s1(p1-c1)^2 = x[n] . w[e]
//   x = [p0^2, p1^2, p0, p1, 1, 0,0,0],  w = [s0, s1, -2s0c0, -2s1c1, s0c0^2+s1c1^2, 0,0,0]
// response = exp(-q) reduced per sorted segment with a 3-level reduction:
// registers -> LDS ds_add_f32 (block) -> global_atomic_add_f32 (device).

typedef __attribute__((ext_vector_type(2))) float v2f;
typedef __attribute__((ext_vector_type(8))) float v8f;

#define E_CENTERS   512
#define FEAT        8
#define PTS_PER_WAVE 16
#define WAVES_PER_BLOCK 8
#define PTS_PER_BLOCK (PTS_PER_WAVE * WAVES_PER_BLOCK)   // 128
#define ETILES      (E_CENTERS / 16)                      // 32

// ---------------------------------------------------------------------------
// Prep: w[e] = [s0, s1, -2 s0 c0, -2 s1 c1, s0 c0^2 + s1 c1^2, 0, 0, 0]
// with s_d = softplus(log_sharpness[e,d]) + 1e-6.
// ---------------------------------------------------------------------------
__global__ void slayer_prep_kernel(const float* __restrict__ centers,
                                   const float* __restrict__ log_sharp,
                                   float* __restrict__ w) {
    int e = blockIdx.x * blockDim.x + threadIdx.x;
    if (e >= E_CENTERS) return;
    float c0 = centers[e * 2 + 0];
    float c1 = centers[e * 2 + 1];
    float s0 = log1pf(__expf(log_sharp[e * 2 + 0])) + 1e-6f;
    float s1 = log1pf(__expf(log_sharp[e * 2 + 1])) + 1e-6f;
    float* we = w + e * FEAT;
    we[0] = s0;
    we[1] = s1;
    we[2] = -2.0f * s0 * c0;
    we[3] = -2.0f * s1 * c1;
    we[4] = s0 * c0 * c0 + s1 * c1 * c1;
    we[5] = 0.0f;
    we[6] = 0.0f;
    we[7] = 0.0f;
}

// ---------------------------------------------------------------------------
// Main: per wave, one 16-point tile x all 512 centers.
// ---------------------------------------------------------------------------
__global__ void __launch_bounds__(256, 4)
slayer_main_kernel(const float* __restrict__ points,
                   const int*   __restrict__ seg_ids,
                   const float* __restrict__ w,
                   float*       __restrict__ out) {
    __shared__ float sW[E_CENTERS * FEAT];   // 16 KB  (B operand table)
    __shared__ int   sSeg[PTS_PER_BLOCK];    // 512 B  (segment ids)
    __shared__ float sAcc[E_CENTERS];        // 2 KB   (block-level accumulator)

    const int tid    = threadIdx.x;
    const int waveId = tid >> 5;
    const int lane   = tid & 31;
    const int mLane  = lane & 15;       // row (A) / col (B,C) within tile
    const int half   = lane >> 4;       // K/M half selector per ISA layout

    // Stage the 512x8 w table into LDS (1024 float4 loads across 256 threads).
    {
        const float4* src = (const float4*)w;
        float4*       dst = (float4*)sW;
        #pragma unroll
        for (int i = 0; i < 4; ++i)
            dst[tid + 256 * i] = src[tid + 256 * i];
    }
    // Stage the block's 128 segment ids; zero the block accumulator.
    const int blockBase = blockIdx.x * PTS_PER_BLOCK;
    if (tid < PTS_PER_BLOCK)
        sSeg[tid] = seg_ids[blockBase + tid];
    sAcc[tid]       = 0.0f;
    sAcc[tid + 256] = 0.0f;
    __syncthreads();

    // Block is single-segment for ~89% of blocks (sorted ids, ~1024 pts/seg).
    const bool blockUniform = (sSeg[0] == sSeg[PTS_PER_BLOCK - 1]);

    // ---- A operand: features of this wave's 16 points --------------------
    // f32 A 16x4 layout: lanes 0-15 => M=lane, VGPR j holds K=j+2*half.
    // K block 0..3 -> a_lo, K block 4..7 -> a_hi.
    const int rowBase = blockBase + waveId * PTS_PER_WAVE;
    const float2 p = *(const float2*)(points + (size_t)(rowBase + mLane) * 2);

    v2f a_lo, a_hi;
    if (half == 0) {                       // K = 0,1  /  4,5
        a_lo[0] = p.x * p.x;  a_lo[1] = p.y * p.y;
        a_hi[0] = 1.0f;       a_hi[1] = 0.0f;
    } else {                               // K = 2,3  /  6,7
        a_lo[0] = p.x;        a_lo[1] = p.y;
        a_hi[0] = 0.0f;       a_hi[1] = 0.0f;
    }

    // Segment ids for the C rows this lane owns: m = half*8 + i.
    const int wbase = waveId * PTS_PER_WAVE;
    int segv[8];
    #pragma unroll
    for (int i = 0; i < 8; ++i)
        segv[i] = sSeg[wbase + half * 8 + i];
    const int  segFirst   = sSeg[wbase];
    const bool uniformSeg = (segFirst == sSeg[wbase + 15]);   // sorted ids

    // ---- sweep the 32 column tiles of E ----------------------------------
    for (int et = 0; et < ETILES; ++et) {
        const int e = et * 16 + mLane;     // this lane's B column
        const float* we = sW + e * FEAT;
        const v2f b_lo = *(const v2f*)(we + 2 * half);       // K = j+2*half
        const v2f b_hi = *(const v2f*)(we + 4 + 2 * half);

        v8f c = {};
        c = __builtin_amdgcn_wmma_f32_16x16x4_f32(false, a_lo, false, b_lo,
                                                  (short)0, c, false, false);
        c = __builtin_amdgcn_wmma_f32_16x16x4_f32(false, a_hi, false, b_hi,
                                                  (short)0, c, false, false);

        // response = exp(-q); v_exp_f32 (TRANS) co-executes with WMMA.
        float r[8];
        #pragma unroll
        for (int i = 0; i < 8; ++i)
            r[i] = __expf(-c[i]);

        if (uniformSeg) {
            // Whole 16-row tile is one segment: column-sum in registers,
            // fold the two M-halves across lanes with one shuffle.
            float colSum = 0.0f;
            #pragma unroll
            for (int i = 0; i < 8; ++i)
                colSum += r[i];
            colSum += __shfl_xor(colSum, 16);
            if (lane < 16) {
                const int col = et * 16 + lane;
                if (blockUniform) {
                    // Intra-WGP reduction: ds_add_f32, no L2 round trip.
                    atomicAdd(&sAcc[col], colSum);
                } else {
                    atomicAdd(&out[(size_t)segFirst * E_CENTERS + col], colSum);
                }
            }
        } else {
            // Rare boundary tile: per-element global atomics.
            #pragma unroll
            for (int i = 0; i < 8; ++i)
                atomicAdd(&out[(size_t)segv[i] * E_CENTERS + et * 16 + mLane],
                          r[i]);
        }
    }

    // ---- commit block accumulator (uniform blocks only) ------------------
    __syncthreads();
    if (blockUniform) {
        const int seg = sSeg[0];
        float* dst = out + (size_t)seg * E_CENTERS;
        atomicAdd(&dst[tid],       sAcc[tid]);
        atomicAdd(&dst[tid + 256], sAcc[tid + 256]);
    }
}

// ---------------------------------------------------------------------------
extern "C" void kernel_launch(void* const* d_in, const int* in_sizes, int n_in,
                              void* d_out, int out_size, void* d_ws, size_t ws_size,
                              hipStream_t stream) {
    const float* points    = (const float*)d_in[0];   // [N,2] f32
    const int*   seg_ids   = (const int*)  d_in[1];   // [N]   i32 (sorted)
    const float* centers   = (const float*)d_in[2];   // [512,2] f32
    const float* log_sharp = (const float*)d_in[3];   // [512,2] f32

    float* out = (float*)d_out;                       // [256,512] f32
    float* w   = (float*)d_ws;                        // [512,8]   f32 scratch

    const int N = in_sizes[0] / 2;

    // Output is accumulated with atomics: zero it every call (capture-safe).
    hipMemsetAsync(d_out, 0, (size_t)out_size * sizeof(float), stream);

    slayer_prep_kernel<<<(E_CENTERS + 255) / 256, 256, 0, stream>>>(
        centers, log_sharp, w);

    const int blocks = N / PTS_PER_BLOCK;             // 262144/128 = 2048
    slayer_main_kernel<<<blocks, 256, 0, stream>>>(points, seg_ids, w, out);
}